// PointerGeneratorOut_74861279969873
// MI455X (gfx1250) — compile-verified
//
#include <hip/hip_runtime.h>
#include <hip/hip_bf16.h>
#include <math.h>

// Problem constants (match reference setup_inputs)
#define BATCH   1024
#define DDIM    512
#define SLEN    200
#define VG      50000
#define OUTSZ   50000

typedef float v2f __attribute__((ext_vector_type(2)));
typedef float v8f __attribute__((ext_vector_type(8)));

// ---------------------------------------------------------------------------
// Kernel 1: inverse permutation  inv[gen_to_out[j]] = j
// ---------------------------------------------------------------------------
__global__ void invperm_kernel(const int* __restrict__ g2o, int* __restrict__ inv) {
    int j = blockIdx.x * blockDim.x + threadIdx.x;
    if (j < VG) inv[g2o[j]] = j;
}

// ---------------------------------------------------------------------------
// Kernel 2: interp[b] = sigmoid(x[b,:] . Wp + bp)   (one 256-thread block/row)
// ---------------------------------------------------------------------------
__global__ __launch_bounds__(256) void interp_kernel(
    const float* __restrict__ x, const float* __restrict__ Wp,
    const float* __restrict__ bp, float* __restrict__ interp)
{
    __shared__ float red[8];
    const int b   = blockIdx.x;
    const int tid = threadIdx.x;
    float s = 0.f;
    for (int i = tid; i < DDIM; i += 256) s += x[(size_t)b * DDIM + i] * Wp[i];
    // wave reduce (wave32)
    for (int off = 16; off > 0; off >>= 1) s += __shfl_xor(s, off, 32);
    if ((tid & 31) == 0) red[tid >> 5] = s;
    __syncthreads();
    if (tid == 0) {
        float t = 0.f;
        for (int w = 0; w < 8; ++w) t += red[w];
        t += bp[0];
        interp[b] = 1.f / (1.f + __expf(-t));
    }
}

// ---------------------------------------------------------------------------
// Kernel 3: logits L[b,j] = x[b,:] @ Wg[:,j] + bg[j]  via V_WMMA_F32_16X16X4_F32
// Block: 256 threads = 8 waves, each wave owns one 16-column strip (N) and
// 4 stacked 16-row M tiles -> block tile 64M x 128N. Each B fragment (global)
// feeds 4 WMMAs, cutting Wg re-read traffic 4x vs one-tile-per-wave.
// A tile (64x512) staged in LDS with pitch padding against bank conflicts.
// ---------------------------------------------------------------------------
#define GEMM_BM   64
#define GEMM_BN   128
#define MT        4      // M tiles per wave
#define A_PITCH   516    // 512 + 4 floats -> rows land on distinct banks

__global__ __launch_bounds__(256) void gemm_logits_kernel(
    const float* __restrict__ x, const float* __restrict__ Wg,
    const float* __restrict__ bg, float* __restrict__ L)
{
    __shared__ float sA[GEMM_BM * A_PITCH];   // 64*516*4 = 132096 B

    const int tid  = threadIdx.x;
    const int row0 = blockIdx.y * GEMM_BM;
    const int col0 = blockIdx.x * GEMM_BN;

    // ---- stage A tile: 64 rows x 512 cols = 8192 float4 / 256 threads ----
    {
        const float4* xv = (const float4*)(x + (size_t)row0 * DDIM);
#pragma unroll
        for (int i = 0; i < 32; ++i) {
            int idx = tid + i * 256;
            int r   = idx >> 7;      // /128 float4 per row
            int c4  = idx & 127;
            float4 v = xv[r * 128 + c4];
            float* dst = &sA[r * A_PITCH + c4 * 4];
            dst[0] = v.x; dst[1] = v.y; dst[2] = v.z; dst[3] = v.w;
        }
    }
    __syncthreads();

    const int wave = tid >> 5;           // 0..7 -> N strip
    const int lane = tid & 31;
    const int m    = lane & 15;          // A row within tile / B-C col
    const int kh   = lane >> 4;          // K half-group select

    const int colBase  = col0 + wave * 16 + m;
    const int colClamp = colBase < VG ? colBase : VG - 1;   // keep loads legal

    const float* __restrict__ sAbase = &sA[m * A_PITCH];

    v8f c0 = {0.f,0.f,0.f,0.f,0.f,0.f,0.f,0.f};
    v8f c1 = c0, c2 = c0, c3 = c0;

#pragma unroll 4
    for (int kk = 0; kk < DDIM; kk += 4) {
        // B fragment (shared by all 4 M tiles):
        //   lane holds Wg[kk+2*kh][col], Wg[kk+2*kh+1][col]
        v2f bfrag;
        const float* bp0 = Wg + (size_t)(kk + 2 * kh) * VG + colClamp;
        bfrag.x = bp0[0];
        bfrag.y = bp0[VG];

        // A fragments: lane holds A[m][2*kh], A[m][2*kh+1] for each M tile
        const int ko = kk + 2 * kh;
        v2f a0, a1, a2, a3;
        a0.x = sAbase[ 0 * 16 * A_PITCH + ko]; a0.y = sAbase[ 0 * 16 * A_PITCH + ko + 1];
        a1.x = sAbase[ 1 * 16 * A_PITCH + ko]; a1.y = sAbase[ 1 * 16 * A_PITCH + ko + 1];
        a2.x = sAbase[ 2 * 16 * A_PITCH + ko]; a2.y = sAbase[ 2 * 16 * A_PITCH + ko + 1];
        a3.x = sAbase[ 3 * 16 * A_PITCH + ko]; a3.y = sAbase[ 3 * 16 * A_PITCH + ko + 1];

        c0 = __builtin_amdgcn_wmma_f32_16x16x4_f32(false, a0, false, bfrag, (short)0, c0, false, false);
        c1 = __builtin_amdgcn_wmma_f32_16x16x4_f32(false, a1, false, bfrag, (short)0, c1, false, false);
        c2 = __builtin_amdgcn_wmma_f32_16x16x4_f32(false, a2, false, bfrag, (short)0, c2, false, false);
        c3 = __builtin_amdgcn_wmma_f32_16x16x4_f32(false, a3, false, bfrag, (short)0, c3, false, false);
    }

    // C layout: VGPR r holds row (r + 8*kh), col = lane&15
    if (colBase < VG) {
        float bias = bg[colBase];
        v8f acc[MT] = {c0, c1, c2, c3};
#pragma unroll
        for (int t = 0; t < MT; ++t) {
            int rbase = row0 + t * 16 + 8 * kh;
#pragma unroll
            for (int r = 0; r < 8; ++r) {
                L[(size_t)(rbase + r) * VG + colBase] = acc[t][r] + bias;
            }
        }
    }
}

// ---------------------------------------------------------------------------
// Kernel 4: fused per-row softmax + permutation remap + pointer scatter + log.
// One block per row; whole 50000-float row staged in LDS (195 KB < 320 KB).
// Reads logits from L (== d_out) fully into LDS before overwriting d_out row.
// ---------------------------------------------------------------------------
__global__ __launch_bounds__(1024) void finalize_kernel(
    const float* __restrict__ L, const float* __restrict__ alphas,
    const int*  __restrict__ ctx_inp, const int* __restrict__ inp_to_out,
    const int*  __restrict__ inv, const float* __restrict__ interp,
    float* __restrict__ out)
{
    __shared__ float vals[VG];      // 200000 B
    __shared__ float red[32];
    __shared__ float bcast;

    const int b    = blockIdx.x;
    const int tid  = threadIdx.x;
    const int lane = tid & 31;
    const int wid  = tid >> 5;

    const float* __restrict__ Lrow = L + (size_t)b * VG;

    // 1) stage full logits row into LDS
    for (int j = tid; j < VG; j += 1024) vals[j] = Lrow[j];
    __syncthreads();

    // 2) block max
    float mx = -INFINITY;
    for (int j = tid; j < VG; j += 1024) mx = fmaxf(mx, vals[j]);
    for (int off = 16; off > 0; off >>= 1) mx = fmaxf(mx, __shfl_xor(mx, off, 32));
    if (lane == 0) red[wid] = mx;
    __syncthreads();
    if (tid < 32) {
        float v = red[tid];
        for (int off = 16; off > 0; off >>= 1) v = fmaxf(v, __shfl_xor(v, off, 32));
        if (tid == 0) bcast = v;
    }
    __syncthreads();
    mx = bcast;
    __syncthreads();

    // 3) block sum of exp
    float sum = 0.f;
    for (int j = tid; j < VG; j += 1024) sum += __expf(vals[j] - mx);
    for (int off = 16; off > 0; off >>= 1) sum += __shfl_xor(sum, off, 32);
    if (lane == 0) red[wid] = sum;
    __syncthreads();
    if (tid < 32) {
        float v = red[tid];
        for (int off = 16; off > 0; off >>= 1) v += __shfl_xor(v, off, 32);
        if (tid == 0) bcast = v;
    }
    __syncthreads();
    const float Z     = bcast;
    const float itp   = interp[b];
    const float scale = itp / Z;

    // 4) convert LDS in place: vals[j] = interp * softmax_j
    for (int j = tid; j < VG; j += 1024) vals[j] = __expf(vals[j] - mx) * scale;
    __syncthreads();

    // 5) pointer scatter into gen-index space: vals[inv[o]] += (1-itp)*alpha
    const float onem = 1.f - itp;
    for (int s = tid; s < SLEN; s += 1024) {
        int o = inp_to_out[ctx_inp[(size_t)b * SLEN + s]];
        atomicAdd(&vals[inv[o]], onem * alphas[(size_t)b * SLEN + s]);
    }
    __syncthreads();

    // 6) coalesced log-prob output: out[b,o] = log(vals[inv[o]])
    float* __restrict__ orow = out + (size_t)b * OUTSZ;
    for (int o = tid; o < OUTSZ; o += 1024) orow[o] = __logf(vals[inv[o]]);
}

// ---------------------------------------------------------------------------
// Launch
// ---------------------------------------------------------------------------
extern "C" void kernel_launch(void* const* d_in, const int* in_sizes, int n_in,
                              void* d_out, int out_size, void* d_ws, size_t ws_size,
                              hipStream_t stream) {
    const float* x          = (const float*)d_in[0];   // [1024,512]
    const float* alphas     = (const float*)d_in[1];   // [1024,200]
    const int*   ctx_inp    = (const int*)  d_in[2];   // [1024,200]
    const int*   gen_to_out = (const int*)  d_in[3];   // [50000]
    const int*   inp_to_out = (const int*)  d_in[4];   // [20000]
    const float* Wp         = (const float*)d_in[5];   // [512]
    const float* bp         = (const float*)d_in[6];   // [1]
    const float* Wg         = (const float*)d_in[7];   // [512,50000]
    const float* bg         = (const float*)d_in[8];   // [50000]

    float* out = (float*)d_out;                        // [1024,50000]

    // workspace layout: interp[1024] f32 | inv[50000] i32
    float* interp = (float*)d_ws;
    int*   inv    = (int*)((char*)d_ws + 4096);

    // 1) inverse permutation
    invperm_kernel<<<(VG + 255) / 256, 256, 0, stream>>>(gen_to_out, inv);

    // 2) mixing probability per row
    interp_kernel<<<BATCH, 256, 0, stream>>>(x, Wp, bp, interp);

    // 3) WMMA GEMM: logits into d_out (used as scratch, same size)
    dim3 ggrid((VG + GEMM_BN - 1) / GEMM_BN, BATCH / GEMM_BM);
    gemm_logits_kernel<<<ggrid, 256, 0, stream>>>(x, Wg, bg, out);

    // 4) fused softmax + remap + pointer + log (in-place over d_out)
    finalize_kernel<<<BATCH, 1024, 0, stream>>>(out, alphas, ctx_inp, inp_to_out,
                                                inv, interp, out);
}